// GNN_76897094468147
// MI455X (gfx1250) — compile-verified
//
#include <hip/hip_runtime.h>
#include <hip/hip_bf16.h>

// ---------------------------------------------------------------------------
// YOLO-style post-processing on gfx1250 (MI455X), wave32.
//  K1: score/class per candidate (memory-bound streaming, ~100MB)
//  K2: exact top-512 per image (tournament selection, shfl_xor wave32 reduce)
//  K3: greedy NMS (sequential, LDS) + exact top-300
//  K4: ROI-align 1x1 over 4 pyramid levels -> f (B x 304 x 1920), M padded
//  K5: h1 = leaky(f @ W1 + b1)    via V_WMMA_F32_16X16X4_F32 (exact fp32)
//  K6: h2 = leaky(h1 @ W2 + b2)   via WMMA; assemble out = [bbox/768, h2]*mask
// ---------------------------------------------------------------------------

#define NB 8
#define NCAND 36720
#define NCLS 80
#define KC 512
#define MDET 300
#define CONF_T 0.596f
#define IOU_T 0.45f
#define MAXWH 4096.0f
#define IMGF 768.0f
#define MPAD 304          // 300 rounded up to multiple of 16
#define KDIM 1920
#define MT 19             // MPAD / 16

typedef float v2f __attribute__((ext_vector_type(2)));
typedef float v8f __attribute__((ext_vector_type(8)));

__device__ __forceinline__ unsigned ordf(float f) {
  unsigned u = __float_as_uint(f);
  return (u & 0x80000000u) ? ~u : (u | 0x80000000u);
}
__device__ __forceinline__ float unordf(unsigned u) {
  return (u & 0x80000000u) ? __uint_as_float(u & 0x7FFFFFFFu)
                           : __uint_as_float(~u);
}
__device__ __forceinline__ unsigned long long shflxor64(unsigned long long v, int m) {
  unsigned lo = (unsigned)v, hi = (unsigned)(v >> 32);
  lo = __shfl_xor(lo, m, 32);
  hi = __shfl_xor(hi, m, 32);
  return ((unsigned long long)hi << 32) | lo;
}
__device__ __forceinline__ unsigned long long wave_max64(unsigned long long k) {
  #pragma unroll
  for (int m = 16; m >= 1; m >>= 1) {
    unsigned long long o = shflxor64(k, m);
    if (o > k) k = o;
  }
  return k;
}
__device__ __forceinline__ int imin(int a, int b) { return a < b ? a : b; }

// ---------------- K1: scores ------------------------------------------------
__global__ __launch_bounds__(256) void score_kernel(
    const float* __restrict__ preds, float* __restrict__ scoreB,
    int* __restrict__ clsB) {
  size_t row = (size_t)blockIdx.x * 256 + threadIdx.x;
  if (row >= (size_t)NB * NCAND) return;
  const float* p = preds + row * 85;
  float obj = p[4];
  float best = p[5] * obj;
  int bi = 0;
  for (int c = 1; c < NCLS; ++c) {
    float v = p[5 + c] * obj;
    if (v > best) { best = v; bi = c; }
  }
  scoreB[row] = (obj > CONF_T && best > CONF_T) ? best : 0.0f;
  clsB[row] = bi;
}

// ---------------- K2: top-512 per image ------------------------------------
// 1024 threads; thread t owns global indices t + k*1024, k in [0,36).
__global__ __launch_bounds__(1024) void topk512_kernel(
    const float* __restrict__ scoreB, float* __restrict__ tsc,
    int* __restrict__ tidx) {
  __shared__ unsigned long long s_red[32];
  __shared__ unsigned long long s_win;
  const int b = blockIdx.x, tid = threadIdx.x;
  const int lane = tid & 31, wid = tid >> 5;
  const float* s = scoreB + (size_t)b * NCAND;

  float lv[36];
  unsigned long long taken = 0ull;
  #pragma unroll
  for (int k = 0; k < 36; ++k) {
    int g = k * 1024 + tid;
    lv[k] = (g < NCAND) ? s[g] : -1.0f;   // sentinel ranks below any score>=0
  }
  auto recompute = [&]() -> unsigned long long {
    unsigned long long best = 0ull;
    #pragma unroll
    for (int k = 0; k < 36; ++k) {
      if (!((taken >> k) & 1ull)) {
        unsigned g = (unsigned)(k * 1024 + tid);
        unsigned long long key =
            ((unsigned long long)ordf(lv[k]) << 32) | (0xFFFFFFFFu - g);
        if (key > best) best = key;
      }
    }
    return best;
  };
  unsigned long long lbest = recompute();

  for (int r = 0; r < KC; ++r) {
    unsigned long long wk = wave_max64(lbest);
    __syncthreads();
    if (lane == 0) s_red[wid] = wk;
    __syncthreads();
    if (wid == 0) {
      unsigned long long kk = s_red[lane];
      kk = wave_max64(kk);
      if (lane == 0) {
        s_win = kk;
        unsigned g = 0xFFFFFFFFu - (unsigned)kk;
        tsc[b * KC + r] = unordf((unsigned)(kk >> 32));
        tidx[b * KC + r] = (int)g;
      }
    }
    __syncthreads();
    unsigned g = 0xFFFFFFFFu - (unsigned)s_win;
    if ((g & 1023u) == (unsigned)tid) {
      taken |= 1ull << (g >> 10);
      lbest = recompute();
    }
  }
}

// ---------------- K3: NMS + top-300 ----------------------------------------
__global__ __launch_bounds__(512) void nms_kernel(
    const float* __restrict__ preds, const float* __restrict__ tsc,
    const int* __restrict__ tidx, const int* __restrict__ clsB,
    float* __restrict__ bboxB, float* __restrict__ maskB) {
  __shared__ float sx1[KC], sy1[KC], sx2[KC], sy2[KC], sarea[KC];
  __shared__ int skeep[KC];
  __shared__ unsigned long long s_red[16];
  __shared__ unsigned long long s_win;
  const int b = blockIdx.x, j = threadIdx.x;
  const int lane = j & 31, wid = j >> 5;

  float sc = tsc[b * KC + j];
  int g = tidx[b * KC + j];
  const float* p = preds + ((size_t)b * NCAND + g) * 85;
  float x = p[0], y = p[1], w = p[2], h = p[3];
  float bx1 = x - 0.5f * w, by1 = y - 0.5f * h;
  float bx2 = x + 0.5f * w, by2 = y + 0.5f * h;
  float off = (float)clsB[(size_t)b * NCAND + g] * MAXWH;
  sx1[j] = bx1 + off; sy1[j] = by1 + off;
  sx2[j] = bx2 + off; sy2[j] = by2 + off;
  sarea[j] = (bx2 - bx1) * (by2 - by1);
  skeep[j] = (sc > 0.0f) ? 1 : 0;
  __syncthreads();

  // greedy sequential NMS, matches fori_loop semantics (dynamic keep[i])
  for (int i = 0; i < KC; ++i) {
    if (skeep[i] && j > i && skeep[j]) {
      float xx1 = fmaxf(sx1[i], sx1[j]);
      float yy1 = fmaxf(sy1[i], sy1[j]);
      float xx2 = fminf(sx2[i], sx2[j]);
      float yy2 = fminf(sy2[i], sy2[j]);
      float iw = fmaxf(xx2 - xx1, 0.0f);
      float ih = fmaxf(yy2 - yy1, 0.0f);
      float inter = iw * ih;
      float iou = inter / (sarea[i] + sarea[j] - inter + 1e-7f);
      if (iou > IOU_T) skeep[j] = 0;
    }
    __syncthreads();
  }

  float sc2 = skeep[j] ? sc : 0.0f;
  bool taken = false;
  for (int r = 0; r < MDET; ++r) {
    unsigned long long key =
        taken ? 0ull
              : (((unsigned long long)ordf(sc2) << 32) |
                 (0xFFFFFFFFu - (unsigned)j));
    unsigned long long wk = wave_max64(key);
    __syncthreads();
    if (lane == 0) s_red[wid] = wk;
    __syncthreads();
    if (wid == 0) {
      unsigned long long kk = (lane < 16) ? s_red[lane] : 0ull;
      kk = wave_max64(kk);
      if (lane == 0) s_win = kk;
    }
    __syncthreads();
    unsigned slot = 0xFFFFFFFFu - (unsigned)s_win;
    if ((unsigned)j == slot) {
      taken = true;
      float* bo = bboxB + ((size_t)b * MDET + r) * 4;
      bo[0] = bx1; bo[1] = by1; bo[2] = bx2; bo[3] = by2;
      maskB[b * MDET + r] = (sc2 > 0.0f) ? 1.0f : 0.0f;
    }
  }
}

// ---------------- K4: ROI-align 1x1 ----------------------------------------
__global__ __launch_bounds__(256) void roi_kernel(
    const float* __restrict__ f1, const float* __restrict__ f2,
    const float* __restrict__ f3, const float* __restrict__ f4,
    const float* __restrict__ bboxB, float* __restrict__ fB) {
  const int r = blockIdx.x % MPAD, b = blockIdx.x / MPAD;
  float* out = fB + ((size_t)b * MPAD + r) * KDIM;
  if (r >= MDET) {
    for (int c = threadIdx.x; c < KDIM; c += 256) out[c] = 0.0f;
    return;
  }
  const float* bo = bboxB + ((size_t)b * MDET + r) * 4;
  const float qx1 = bo[0], qy1 = bo[1], qx2 = bo[2], qy2 = bo[3];

  const float* feats[4] = {f1, f2, f3, f4};
  const int Cs[4] = {128, 256, 512, 1024};
  const int HWs[4] = {96, 48, 24, 12};
  const float scl[4] = {0.125f, 0.0625f, 0.03125f, 0.015625f};
  const int cb[4] = {0, 128, 384, 896};

  #pragma unroll
  for (int l = 0; l < 4; ++l) {
    const int C = Cs[l], H = HWs[l], W = HWs[l];
    const float s = scl[l];
    float x1 = qx1 * s, y1 = qy1 * s, x2 = qx2 * s, y2 = qy2 * s;
    float rw = fmaxf(x2 - x1, 1.0f), rh = fmaxf(y2 - y1, 1.0f);
    float ysv[2] = {y1 + rh * 0.25f, y1 + rh * 0.75f};
    float xsv[2] = {x1 + rw * 0.25f, x1 + rw * 0.75f};
    int o00[4], o01[4], o10[4], o11[4];
    float w00[4], w01[4], w10[4], w11[4];
    bool val[4];
    #pragma unroll
    for (int sp = 0; sp < 4; ++sp) {
      float yy = ysv[sp >> 1], xx = xsv[sp & 1];
      bool v = (yy > -1.0f) && (yy < (float)H) && (xx > -1.0f) && (xx < (float)W);
      float yc = fmaxf(yy, 0.0f), xc = fmaxf(xx, 0.0f);
      int y0 = (int)fminf(floorf(yc), (float)(H - 1));
      int x0 = (int)fminf(floorf(xc), (float)(W - 1));
      int y1i = imin(y0 + 1, H - 1), x1i = imin(x0 + 1, W - 1);
      float ly = yc - (float)y0, lx = xc - (float)x0;
      float hy = 1.0f - ly, hx = 1.0f - lx;
      val[sp] = v;
      o00[sp] = y0 * W + x0;  o01[sp] = y0 * W + x1i;
      o10[sp] = y1i * W + x0; o11[sp] = y1i * W + x1i;
      w00[sp] = hy * hx; w01[sp] = hy * lx;
      w10[sp] = ly * hx; w11[sp] = ly * lx;
    }
    const float* fbase = feats[l] + (size_t)b * C * H * W;
    for (int c = threadIdx.x; c < C; c += 256) {
      const float* fp = fbase + (size_t)c * H * W;
      float acc = 0.0f;
      #pragma unroll
      for (int sp = 0; sp < 4; ++sp) {
        if (val[sp]) {
          acc += w00[sp] * fp[o00[sp]] + w01[sp] * fp[o01[sp]] +
                 w10[sp] * fp[o10[sp]] + w11[sp] * fp[o11[sp]];
        }
      }
      out[cb[l] + c] = acc * 0.25f;
    }
  }
}

// ---------------- K5: GEMM1 (f @ W1 + b1, leaky) ---------------------------
// 4 waves/block; wave w owns N-tile w; block = one M-tile of one image.
__global__ __launch_bounds__(128) void gemm1_kernel(
    const float* __restrict__ fB, const float* __restrict__ W1,
    const float* __restrict__ b1, float* __restrict__ h1B) {
  const int b = blockIdx.x / MT, mt = blockIdx.x % MT;
  const int wv = threadIdx.x >> 5, lane = threadIdx.x & 31;
  const int m = mt * 16 + (lane & 15);
  const int n = wv * 16 + (lane & 15);
  const int koff = (lane >> 4) * 2;
  const float* arow = fB + ((size_t)b * MPAD + m) * KDIM;
  v8f c = {};
  for (int kb = 0; kb < KDIM; kb += 4) {
    v2f a, bb;
    a.x = arow[kb + koff];
    a.y = arow[kb + koff + 1];
    bb.x = W1[(size_t)(kb + koff) * 64 + n];
    bb.y = W1[(size_t)(kb + koff + 1) * 64 + n];
    c = __builtin_amdgcn_wmma_f32_16x16x4_f32(false, a, false, bb,
                                              (short)0, c, false, false);
  }
  float bias = b1[n];
  #pragma unroll
  for (int r = 0; r < 8; ++r) {
    int row = mt * 16 + r + ((lane >> 4) * 8);
    float v = c[r] + bias;
    v = v > 0.0f ? v : 0.01f * v;
    h1B[((size_t)b * MPAD + row) * 64 + n] = v;
  }
}

// ---------------- K6: GEMM2 + output assembly ------------------------------
__global__ __launch_bounds__(128) void gemm2_kernel(
    const float* __restrict__ h1B, const float* __restrict__ W2,
    const float* __restrict__ b2, const float* __restrict__ bboxB,
    const float* __restrict__ maskB, float* __restrict__ out) {
  const int b = blockIdx.x / MT, mt = blockIdx.x % MT;
  const int wv = threadIdx.x >> 5, lane = threadIdx.x & 31;
  const int m = mt * 16 + (lane & 15);
  const int n = wv * 16 + (lane & 15);
  const int koff = (lane >> 4) * 2;
  const float* arow = h1B + ((size_t)b * MPAD + m) * 64;
  v8f c = {};
  #pragma unroll
  for (int kb = 0; kb < 64; kb += 4) {
    v2f a, bb;
    a.x = arow[kb + koff];
    a.y = arow[kb + koff + 1];
    bb.x = W2[(kb + koff) * 64 + n];
    bb.y = W2[(kb + koff + 1) * 64 + n];
    c = __builtin_amdgcn_wmma_f32_16x16x4_f32(false, a, false, bb,
                                              (short)0, c, false, false);
  }
  float bias = b2[n];
  #pragma unroll
  for (int r = 0; r < 8; ++r) {
    int row = mt * 16 + r + ((lane >> 4) * 8);
    if (row < MDET) {
      float mk = maskB[b * MDET + row];
      float v = c[r] + bias;
      v = v > 0.0f ? v : 0.01f * v;
      out[((size_t)b * MDET + row) * 68 + 4 + n] = v * mk;
    }
  }
  if (threadIdx.x < 64) {
    int rl = threadIdx.x >> 2, col = threadIdx.x & 3;
    int row = mt * 16 + rl;
    if (row < MDET) {
      float mk = maskB[b * MDET + row];
      out[((size_t)b * MDET + row) * 68 + col] =
          bboxB[((size_t)b * MDET + row) * 4 + col] * mk * (1.0f / IMGF);
    }
  }
}

// ---------------------------------------------------------------------------
extern "C" void kernel_launch(void* const* d_in, const int* in_sizes, int n_in,
                              void* d_out, int out_size, void* d_ws,
                              size_t ws_size, hipStream_t stream) {
  const float* preds = (const float*)d_in[0];
  const float* f1 = (const float*)d_in[1];
  const float* f2 = (const float*)d_in[2];
  const float* f3 = (const float*)d_in[3];
  const float* f4 = (const float*)d_in[4];
  const float* W1 = (const float*)d_in[5];
  const float* b1 = (const float*)d_in[6];
  const float* W2 = (const float*)d_in[7];
  const float* b2 = (const float*)d_in[8];
  float* out = (float*)d_out;

  float* ws = (float*)d_ws;
  size_t o = 0;
  float* scoreB = ws + o; o += (size_t)NB * NCAND;
  int* clsB = (int*)(ws + o); o += (size_t)NB * NCAND;
  float* tsc = ws + o; o += (size_t)NB * KC;
  int* tidx = (int*)(ws + o); o += (size_t)NB * KC;
  float* bboxB = ws + o; o += (size_t)NB * MDET * 4;
  float* maskB = ws + o; o += (size_t)NB * MDET;
  float* fB = ws + o; o += (size_t)NB * MPAD * KDIM;
  float* h1B = ws + o; o += (size_t)NB * MPAD * 64;

  const int rows = NB * NCAND;
  score_kernel<<<(rows + 255) / 256, 256, 0, stream>>>(preds, scoreB, clsB);
  topk512_kernel<<<NB, 1024, 0, stream>>>(scoreB, tsc, tidx);
  nms_kernel<<<NB, KC, 0, stream>>>(preds, tsc, tidx, clsB, bboxB, maskB);
  roi_kernel<<<NB * MPAD, 256, 0, stream>>>(f1, f2, f3, f4, bboxB, fB);
  gemm1_kernel<<<NB * MT, 128, 0, stream>>>(fB, W1, b1, h1B);
  gemm2_kernel<<<NB * MT, 128, 0, stream>>>(h1B, W2, b2, bboxB, maskB, out);
}